// GCN_19679540150348
// MI455X (gfx1250) — compile-verified
//
#include <hip/hip_runtime.h>

typedef __attribute__((ext_vector_type(2))) float v2f;
typedef __attribute__((ext_vector_type(8))) float v8f;

#define FEAT_DIM 64

// ---------------------------------------------------------------------------
// Zero-fill (graph-capture-safe replacement for memset; d_ws is poisoned 0xAA)
// ---------------------------------------------------------------------------
__global__ void fill_zero_kernel(float* __restrict__ p, long long n) {
  long long i = (long long)blockIdx.x * blockDim.x + threadIdx.x;
  if (i < n) p[i] = 0.0f;
}

// ---------------------------------------------------------------------------
// Scatter-add SpMM: agg[dst[e]][:] += feat[src[e]][:] * norm[src[e]]
// One wave32 per edge; each lane moves a float2 (64 floats per edge).
// Feature/agg arrays (25.6 MB) fit in the 192 MB L2, so the gathers and the
// f32 atomics resolve at L2 scope.
// ---------------------------------------------------------------------------
__global__ void scatter_add_scaled(const float* __restrict__ feat,
                                   const float* __restrict__ norm,
                                   const int* __restrict__ src,
                                   const int* __restrict__ dst,
                                   float* __restrict__ agg, int E) {
  int tid = blockIdx.x * blockDim.x + threadIdx.x;
  int e = tid >> 5;
  if (e >= E) return;
  int lane = tid & 31;
  int s = src[e];
  int t = dst[e];
  float ns = norm[s];
  const float2* f2 = reinterpret_cast<const float2*>(feat + (size_t)s * FEAT_DIM);
  float2 v = f2[lane];
  float* ap = agg + (size_t)t * FEAT_DIM + lane * 2;
  atomicAdd(ap + 0, v.x * ns);
  atomicAdd(ap + 1, v.y * ns);
}

// ---------------------------------------------------------------------------
// Dense node-apply via V_WMMA_F32_16X16X4_F32 (full fp32, K=64):
//   out[r][c] = act( sum_k (agg[r][k]*norm[r]) * W[k][c] + b[c] )
// One wave per 16x16 output tile; 16 chained WMMAs over K.
// A layout (ISA 7.12.2, 32-bit A 16x4): lane m = M, half = lane>>4,
//   VGPRs hold K = k0+2*half and k0+2*half+1.
// B layout (4x16, row striped across lanes): VGPR0 = K {k0, k0+2},
//   VGPR1 = K {k0+1, k0+3}, col = lane&15.
// C/D layout: acc[j] -> row j + 8*half, col lane&15.
// ---------------------------------------------------------------------------
template <int NCOLS, int NTILES_N, bool RELU>
__global__ void dense_wmma(const float* __restrict__ agg,
                           const float* __restrict__ norm,
                           const float* __restrict__ W,     // [FEAT_DIM, NCOLS] row-major
                           const float* __restrict__ bias,  // [NCOLS]
                           float* __restrict__ out,         // [nrows, NCOLS]
                           int nrows) {
  const int lane = threadIdx.x & 31;
  const int wave = threadIdx.x >> 5;
  const int wid = blockIdx.x * (blockDim.x >> 5) + wave;
  const int tiles_m = (nrows + 15) >> 4;
  if (wid >= tiles_m * NTILES_N) return;  // uniform per wave: EXEC all-1s for WMMA
  const int tile_m = wid / NTILES_N;
  const int tile_n = wid % NTILES_N;
  const int r0 = tile_m << 4;
  const int c0 = tile_n << 4;

  const int m = lane & 15;
  const int half = lane >> 4;

  int r = r0 + m;
  if (r >= nrows) r = nrows - 1;  // clamp (padded rows never stored)
  const float nr = norm[r];
  const float* __restrict__ arow = agg + (size_t)r * FEAT_DIM;

  const int col = c0 + (lane & 15);
  const bool colok = (col < NCOLS);
  const int bcol = colok ? col : 0;  // branchless guard keeps EXEC full

  v8f acc = {};
#pragma unroll
  for (int k0 = 0; k0 < FEAT_DIM; k0 += 4) {
    const int k = k0 + half * 2;
    v2f a;
    a.x = arow[k] * nr;
    a.y = arow[k + 1] * nr;
    v2f b;
    b.x = W[(size_t)k * NCOLS + bcol];
    b.y = W[(size_t)(k + 1) * NCOLS + bcol];
    // (neg_a, A, neg_b, B, c_mod, C, reuse_a, reuse_b)
    acc = __builtin_amdgcn_wmma_f32_16x16x4_f32(false, a, false, b, (short)0,
                                                acc, false, false);
  }

  if (colok) {
    const float bv = bias[bcol];
#pragma unroll
    for (int j = 0; j < 8; ++j) {
      const int row = r0 + j + half * 8;
      if (row < nrows) {
        float v = acc[j] + bv;
        if (RELU) v = fmaxf(v, 0.0f);
        out[(size_t)row * NCOLS + col] = v;
      }
    }
  }
}

// ---------------------------------------------------------------------------
// Host-side orchestration (all on `stream`; graph-capture safe)
// ---------------------------------------------------------------------------
extern "C" void kernel_launch(void* const* d_in, const int* in_sizes, int n_in,
                              void* d_out, int out_size, void* d_ws, size_t ws_size,
                              hipStream_t stream) {
  const float* feat = (const float*)d_in[0];  // [N,64]
  const float* norm = (const float*)d_in[1];  // [N,1]
  const int* src    = (const int*)d_in[2];    // [E]
  const int* dst    = (const int*)d_in[3];    // [E]
  const float* W1   = (const float*)d_in[4];  // [64,64]
  const float* b1   = (const float*)d_in[5];  // [64]
  const float* W2   = (const float*)d_in[6];  // [64,40]
  const float* b2   = (const float*)d_in[7];  // [40]
  float* out = (float*)d_out;                 // [N,40]

  const int N = in_sizes[1];  // norm is [N,1]
  const int E = in_sizes[2];

  float* bufA = (float*)d_ws;                  // aggregation buffer [N,64]
  float* bufB = bufA + (size_t)N * FEAT_DIM;   // hidden h1         [N,64]

  const long long aggN = (long long)N * FEAT_DIM;
  const int fillBlocks = (int)((aggN + 255) / 256);
  const int scatBlocks = (E + 7) / 8;  // E*32 threads / 256

  const int tilesM = (N + 15) / 16;
  const int waves1 = tilesM * 4;  // 64 output cols -> 4 N-tiles
  const int waves2 = tilesM * 3;  // 40 output cols -> 3 N-tiles (guarded)
  const int dense1Blocks = (waves1 + 7) / 8;  // 8 waves / 256-thread block
  const int dense2Blocks = (waves2 + 7) / 8;

  // ---- Layer 1 ----
  fill_zero_kernel<<<fillBlocks, 256, 0, stream>>>(bufA, aggN);
  scatter_add_scaled<<<scatBlocks, 256, 0, stream>>>(feat, norm, src, dst, bufA, E);
  dense_wmma<64, 4, true><<<dense1Blocks, 256, 0, stream>>>(bufA, norm, W1, b1, bufB, N);

  // ---- Layer 2 ----
  fill_zero_kernel<<<fillBlocks, 256, 0, stream>>>(bufA, aggN);
  scatter_add_scaled<<<scatBlocks, 256, 0, stream>>>(bufB, norm, src, dst, bufA, E);
  dense_wmma<40, 3, false><<<dense2Blocks, 256, 0, stream>>>(bufA, norm, W2, b2, out, N);
}